// GivensLinear_28630251995440
// MI455X (gfx1250) — compile-verified
//
#include <hip/hip_runtime.h>

#define DIM 128
#define NUM_ROT 8128          // 128*127/2
#define CHUNKS 64
#define ROT_PER_CHUNK 127     // 64 * 127 == 8128
#define MAT_ELEMS (DIM * DIM)

typedef float v2f __attribute__((ext_vector_type(2)));
typedef float v8f __attribute__((ext_vector_type(8)));

// D = A(16x4 f32) * B(4x16 f32) + C(16x16 f32), wave32 CDNA5 WMMA
__device__ __forceinline__ v8f wmma_f32_4(v2f a, v2f b, v8f c) {
    return __builtin_amdgcn_wmma_f32_16x16x4_f32(
        /*neg_a=*/false, a, /*neg_b=*/false, b,
        /*c_mod=*/(short)0, c, /*reuse_a=*/false, /*reuse_b=*/false);
}

// ---------------------------------------------------------------------------
// Kernel 1: 64 partial products of 127 rotations each, built in LDS.
// Phase A: threads 0..126 compute cos/sin/(i,j) of this chunk's rotations
//          into LDS (one barrier).
// Phase B: thread t owns matrix column t -> every LDS matrix cell is touched
//          by exactly one thread -> no barriers in the sequential loop.
// ---------------------------------------------------------------------------
__global__ void chunk_kernel(const float* __restrict__ angles,
                             float* __restrict__ chunkMats) {
    extern __shared__ float smem[];
    float* Rch = smem;                       // DIM*DIM floats (64 KB)
    float* csA = smem + MAT_ELEMS;           // 128
    float* snA = csA + DIM;                  // 128
    int*   iA  = (int*)(snA + DIM);          // 128
    int*   jA  = iA + DIM;                   // 128

    const int t = threadIdx.x;               // 0..127, column owner
    const int c = blockIdx.x;                // chunk id
    const int k0 = c * ROT_PER_CHUNK;

    if (t < ROT_PER_CHUNK) {
        const int k = k0 + t;
        const float a = angles[k];
        csA[t] = cosf(a);
        snA[t] = sinf(a);
        int rem = k, i = 0, rowlen = DIM - 1;
        while (rem >= rowlen) { rem -= rowlen; --rowlen; ++i; }
        iA[t] = i;
        jA[t] = i + 1 + rem;
    }

    for (int r = 0; r < DIM; ++r)
        Rch[r * DIM + t] = (r == t) ? 1.0f : 0.0f;

    __syncthreads();   // trig/index tables visible to all columns

    for (int m = 0; m < ROT_PER_CHUNK; ++m) {
        const int i   = iA[m];
        const int j   = jA[m];
        const float cc = csA[m];
        const float ss = snA[m];
        const float ri = Rch[i * DIM + t];
        const float rj = Rch[j * DIM + t];
        Rch[i * DIM + t] = cc * ri - ss * rj;
        Rch[j * DIM + t] = ss * ri + cc * rj;
    }

    float* dst = chunkMats + (size_t)c * MAT_ELEMS;
    for (int r = 0; r < DIM; ++r)
        dst[r * DIM + t] = Rch[r * DIM + t];
}

// ---------------------------------------------------------------------------
// Kernel 2: multiply `group` consecutive 128x128 matrices in application order:
// T = M_0; T = M_t * T (t = 1..group-1). T lives in 64 KB LDS; product done
// with fp32 WMMA. Block = 256 threads = 8 waves; wave w owns rows 16w..16w+15.
// ---------------------------------------------------------------------------
__global__ void combine_kernel(const float* __restrict__ src, float* __restrict__ dst,
                               int group) {
    extern __shared__ float T[];            // DIM*DIM = 64 KB
    const int tid  = threadIdx.x;
    const int wave = tid >> 5;
    const int lane = tid & 31;
    const int l15  = lane & 15;
    const int h    = lane >> 4;

    const float* base = src + (size_t)blockIdx.x * group * MAT_ELEMS;

    for (int e = tid; e < MAT_ELEMS; e += blockDim.x) T[e] = base[e];
    __syncthreads();

    for (int t = 1; t < group; ++t) {
        const float* Mt = base + (size_t)t * MAT_ELEMS;
        v8f acc[8];
        const v8f zero = {};
        #pragma unroll
        for (int nb = 0; nb < 8; ++nb) acc[nb] = zero;

        for (int kb = 0; kb < 32; ++kb) {
            // A fragment: rows of M_t (global, float2 contiguous in K)
            v2f a = *(const v2f*)(Mt + (16 * wave + l15) * DIM + kb * 4 + 2 * h);
            #pragma unroll
            for (int nb = 0; nb < 8; ++nb) {
                v2f b;
                b.x = T[(kb * 4 + 2 * h)     * DIM + nb * 16 + l15];
                b.y = T[(kb * 4 + 2 * h + 1) * DIM + nb * 16 + l15];
                acc[nb] = wmma_f32_4(a, b, acc[nb]);
            }
        }

        __syncthreads();   // everyone done reading old T
        #pragma unroll
        for (int nb = 0; nb < 8; ++nb) {
            #pragma unroll
            for (int r = 0; r < 8; ++r)
                T[(16 * wave + r + 8 * h) * DIM + nb * 16 + l15] = acc[nb][r];
        }
        __syncthreads();   // new T fully written
    }

    float* d = dst + (size_t)blockIdx.x * MAT_ELEMS;
    for (int e = tid; e < MAT_ELEMS; e += blockDim.x) d[e] = T[e];
}

// ---------------------------------------------------------------------------
// Kernel 3: out = X * R^T + bias.  One wave per 16-row slab of X,
// 8 n-tiles x 32 k-steps = 256 fp32 WMMAs per wave.
// X/out are single-touch -> non-temporal; R/bias stay cache-resident (RT).
// ---------------------------------------------------------------------------
__global__ void gemm_kernel(const float* __restrict__ X, const float* __restrict__ R,
                            const float* __restrict__ bias, float* __restrict__ out,
                            int rows) {
    const int tid  = threadIdx.x;
    const int wave = tid >> 5;
    const int lane = tid & 31;
    const int l15  = lane & 15;
    const int h    = lane >> 4;

    const long slab = (long)blockIdx.x * 8 + wave;
    const long r0   = slab * 16;
    if (r0 >= rows) return;                 // wave-uniform; EXEC stays all-1s

    const float* xrow = X + (r0 + l15) * DIM;

    v8f acc[8];
    const v8f zero = {};
    #pragma unroll
    for (int nb = 0; nb < 8; ++nb) acc[nb] = zero;

    for (int kb = 0; kb < 32; ++kb) {
        v2f a = __builtin_nontemporal_load((const v2f*)(xrow + kb * 4 + 2 * h));
        #pragma unroll
        for (int nb = 0; nb < 8; ++nb) {
            // B[k][n] = R[n][k] -> contiguous float2 along k of row n
            v2f b = *(const v2f*)(R + (nb * 16 + l15) * DIM + kb * 4 + 2 * h);
            acc[nb] = wmma_f32_4(a, b, acc[nb]);
        }
    }

    #pragma unroll
    for (int nb = 0; nb < 8; ++nb) {
        const float bv = bias[nb * 16 + l15];
        #pragma unroll
        for (int r = 0; r < 8; ++r) {
            const long row = r0 + r + 8 * h;
            __builtin_nontemporal_store(acc[nb][r] + bv,
                                        out + row * DIM + nb * 16 + l15);
        }
    }
}

// ---------------------------------------------------------------------------
extern "C" void kernel_launch(void* const* d_in, const int* in_sizes, int n_in,
                              void* d_out, int out_size, void* d_ws, size_t ws_size,
                              hipStream_t stream) {
    const float* x      = (const float*)d_in[0];   // (32,4096,128) f32
    const float* angles = (const float*)d_in[1];   // (8128,) f32
    const float* bias   = (const float*)d_in[2];   // (128,) f32
    float* out = (float*)d_out;

    float* ws     = (float*)d_ws;
    float* chunks = ws;                                    // 64 x 128x128
    float* Q      = chunks + (size_t)CHUNKS * MAT_ELEMS;   // 8  x 128x128
    float* Rm     = Q + (size_t)8 * MAT_ELEMS;             // 1  x 128x128

    const int rows = in_sizes[0] / DIM;                    // 131072

    const size_t chunkLds = (MAT_ELEMS + 2 * DIM) * sizeof(float) + 2 * DIM * sizeof(int);
    chunk_kernel<<<CHUNKS, DIM, chunkLds, stream>>>(angles, chunks);
    combine_kernel<<<8, 256, MAT_ELEMS * sizeof(float), stream>>>(chunks, Q, 8);
    combine_kernel<<<1, 256, MAT_ELEMS * sizeof(float), stream>>>(Q, Rm, 8);

    const int blocks = rows / (16 * 8);                    // 1024
    gemm_kernel<<<blocks, 256, 0, stream>>>(x, Rm, bias, out, rows);
}